// ModelNew_25812753449937
// MI455X (gfx1250) — compile-verified
//
#include <hip/hip_runtime.h>
#include <math.h>

typedef __attribute__((ext_vector_type(2))) float v2f;
typedef __attribute__((ext_vector_type(4))) float v4f;
typedef __attribute__((ext_vector_type(8))) float v8f;

#define WMMA_F32(a, b, c) \
  __builtin_amdgcn_wmma_f32_16x16x4_f32(false, (a), false, (b), (short)0, (c), false, false)

static constexpr int BSZ = 4;
static constexpr int LSEQ = 8192;
static constexpr int NH = 16;
static constexpr int NC = 128;   // number of chunks
static constexpr int CH = 64;    // chunk length
static constexpr int LDSS = 68;  // padded LDS row stride (floats) -> conflict-free

// ---- fragment loaders (wave32, f32 16x16x4 WMMA layouts) ----
// A-matrix fragment from row-major LDS tile: A[m, k], m = rm + (lane&15), k = kk + 2*(lane>>4) + v
__device__ __forceinline__ v2f ldA_rowmajor(const float* Sm, int rm, int kk, int half, int l16) {
  return *(const v2f*)&Sm[(rm + l16) * LDSS + kk + 2 * half];
}
// B-matrix fragment where the WMMA-B operand is the TRANSPOSE of a row-major LDS tile:
// B_wmma[k, j] = Sm[j][k]  (lane holds output column j = cn + (lane&15))
__device__ __forceinline__ v2f ldB_transposed(const float* Sm, int cn, int kk, int half, int l16) {
  return *(const v2f*)&Sm[(cn + l16) * LDSS + kk + 2 * half];
}
// B-matrix fragment from a row-major K x N LDS tile: B_wmma[k, j] = Sm[k][j]
__device__ __forceinline__ v2f ldB_rowmajor(const float* Sm, int cn, int kk, int half, int l16) {
  int k0 = kk + 2 * half;
  v2f r;
  r.x = Sm[k0 * LDSS + cn + l16];
  r.y = Sm[(k0 + 1) * LDSS + cn + l16];
  return r;
}

// Parallel inclusive cumsum over cs[0..63] (Hillis-Steele, all 256 threads hit barriers)
__device__ __forceinline__ void cumsum64(float* cs, int tid) {
#pragma unroll
  for (int off = 1; off < 64; off <<= 1) {
    float v = 0.f;
    if (tid < 64) {
      v = cs[tid];
      if (tid >= off) v += cs[tid - off];
    }
    __syncthreads();
    if (tid < 64) cs[tid] = v;
    __syncthreads();
  }
}

// =====================================================================
// Kernel 1: per (b, chunk, head): Y_diag, per-chunk states, chunk totals
// =====================================================================
__global__ __launch_bounds__(256) void ssd_k1_intra(
    const float* __restrict__ X, const float* __restrict__ A,
    const float* __restrict__ Bp, const float* __restrict__ Cp,
    float* __restrict__ Y, float* __restrict__ states, float* __restrict__ Ttot) {
  __shared__ __align__(16) float Cm[64 * LDSS];  // C tile, later aliased as M tile
  __shared__ __align__(16) float Bm[64 * LDSS];
  __shared__ __align__(16) float Xm[64 * LDSS];
  __shared__ float cs[64];
  __shared__ float dec[64];

  const int bid = blockIdx.x;
  const int h = bid % NH;
  const int c = (bid / NH) % NC;
  const int b = bid / (NH * NC);
  const int tid = threadIdx.x;
  const int lane = tid & 31, wv = tid >> 5;
  const int half = lane >> 4, l16 = lane & 15;

  const size_t rstride = (size_t)NH * 64;  // element stride between sequence rows
  const size_t rowbase = ((size_t)(b * LSEQ + c * CH) * NH + h) * 64;

  // A chunk -> inclusive cumsum cs[], decay-to-end dec[], chunk total T
  if (tid < 64) cs[tid] = A[(size_t)(b * LSEQ + c * CH + tid) * NH + h];

  // Stage C, B, X tiles (64x64) into padded LDS via float4
  for (int i = 0; i < 4; ++i) {
    int q = tid + i * 256;
    int r = q >> 4, f4 = (q & 15) * 4;
    size_t g = rowbase + (size_t)r * rstride + f4;
    int ls = r * LDSS + f4;
    *(v4f*)&Cm[ls] = *(const v4f*)&Cp[g];
    *(v4f*)&Bm[ls] = *(const v4f*)&Bp[g];
    *(v4f*)&Xm[ls] = *(const v4f*)&X[g];
  }
  __syncthreads();
  cumsum64(cs, tid);
  const float Tc = cs[63];
  if (tid < 64) dec[tid] = __expf(Tc - cs[tid]);
  if (tid == 0) Ttot[(size_t)(b * NH + h) * NC + c] = Tc;
  __syncthreads();

  // Each wave owns output tiles t0 = 2*wv, t1 = 2*wv+1 (same 16-row block, two col blocks)
  const int t0 = wv * 2;
  const int tr = (t0 >> 2) * 16;
  const int tc0 = (t0 & 3) * 16;
  const int tc1 = tc0 + 16;

  // ---- G = C * B^T ----
  v8f accG0 = {}, accG1 = {};
  for (int kk = 0; kk < 64; kk += 4) {
    v2f a = ldA_rowmajor(Cm, tr, kk, half, l16);
    v2f b0 = ldB_transposed(Bm, tc0, kk, half, l16);
    v2f b1 = ldB_transposed(Bm, tc1, kk, half, l16);
    accG0 = WMMA_F32(a, b0, accG0);
    accG1 = WMMA_F32(a, b1, accG1);
  }
  __syncthreads();  // all reads of Cm complete before overwriting it with M

  // ---- M[l,s] = (l>=s) ? G*exp(cs[l]-cs[s]) : 0, stored over Cm ----
  for (int j = 0; j < 8; ++j) {
    int row = tr + j + 8 * half;
    int col0 = tc0 + l16, col1 = tc1 + l16;
    Cm[row * LDSS + col0] = (row >= col0) ? accG0[j] * __expf(cs[row] - cs[col0]) : 0.0f;
    Cm[row * LDSS + col1] = (row >= col1) ? accG1[j] * __expf(cs[row] - cs[col1]) : 0.0f;
  }
  __syncthreads();

  // ---- Y_diag = M * X  and  states = (X^T * diag(dec)) * B ----
  v8f accY0 = {}, accY1 = {}, accS0 = {}, accS1 = {};
  for (int kk = 0; kk < 64; kk += 4) {
    v2f am = ldA_rowmajor(Cm, tr, kk, half, l16);  // M rows
    v2f bx0 = ldB_rowmajor(Xm, tc0, kk, half, l16);
    v2f bx1 = ldB_rowmajor(Xm, tc1, kk, half, l16);
    accY0 = WMMA_F32(am, bx0, accY0);
    accY1 = WMMA_F32(am, bx1, accY1);

    int k0 = kk + 2 * half;
    v2f ax;  // A[m=p, k=l] = X[l, p] * dec[l]
    ax.x = Xm[k0 * LDSS + tr + l16] * dec[k0];
    ax.y = Xm[(k0 + 1) * LDSS + tr + l16] * dec[k0 + 1];
    v2f bb0 = ldB_rowmajor(Bm, tc0, kk, half, l16);
    v2f bb1 = ldB_rowmajor(Bm, tc1, kk, half, l16);
    accS0 = WMMA_F32(ax, bb0, accS0);
    accS1 = WMMA_F32(ax, bb1, accS1);
  }

  const size_t sbase = ((size_t)((b * NC + c) * NH + h)) * 4096;
  for (int j = 0; j < 8; ++j) {
    int row = tr + j + 8 * half;
    Y[rowbase + (size_t)row * rstride + tc0 + l16] = accY0[j];
    Y[rowbase + (size_t)row * rstride + tc1 + l16] = accY1[j];
    states[sbase + row * 64 + tc0 + l16] = accS0[j];
    states[sbase + row * 64 + tc1 + l16] = accS1[j];
  }
}

// =====================================================================
// Kernel 2: sequential cross-chunk scan per (b,h); converts states ->
// exclusive-scanned S_prev in place:  S_{c+1} = exp(T_c)*S_c + states_c
// Prefetch next chunk's state block to overlap the serial dependence
// chain with memory latency (lowers to global_prefetch_b8).
// =====================================================================
__global__ __launch_bounds__(256) void ssd_k2_scan(
    float* __restrict__ states, const float* __restrict__ Ttot) {
  const int bh = blockIdx.x;  // b*NH + h
  const int b = bh / NH, h = bh % NH;
  const int tid = threadIdx.x;
  const float* tp = &Ttot[(size_t)bh * NC];

  float S[16];
#pragma unroll
  for (int i = 0; i < 16; ++i) S[i] = 0.0f;

  for (int c = 0; c < NC; ++c) {
    float* sp = &states[((size_t)((b * NC + c) * NH + h)) * 4096];
    if (c + 1 < NC) {
      const float* pf = &states[((size_t)((b * NC + c + 1) * NH + h)) * 4096];
#pragma unroll
      for (int i = 0; i < 16; ++i) __builtin_prefetch(&pf[tid + i * 256], 0, 3);
    }
    float e = __expf(tp[c]);
#pragma unroll
    for (int i = 0; i < 16; ++i) {
      int idx = tid + i * 256;
      float v = sp[idx];
      sp[idx] = S[i];          // write state ENTERING chunk c
      S[i] = e * S[i] + v;     // advance past chunk c
    }
  }
}

// =====================================================================
// Kernel 3: Y += exp(cs[l]) * (C * S_prev^T)  per (b, chunk, head)
// =====================================================================
__global__ __launch_bounds__(256) void ssd_k3_off(
    const float* __restrict__ A, const float* __restrict__ Cp,
    const float* __restrict__ states, float* __restrict__ Y) {
  __shared__ __align__(16) float Cm[64 * LDSS];
  __shared__ __align__(16) float Sm[64 * LDSS];  // S_prev[p][n]
  __shared__ float cs[64];

  const int bid = blockIdx.x;
  const int h = bid % NH;
  const int c = (bid / NH) % NC;
  const int b = bid / (NH * NC);
  const int tid = threadIdx.x;
  const int lane = tid & 31, wv = tid >> 5;
  const int half = lane >> 4, l16 = lane & 15;

  const size_t rstride = (size_t)NH * 64;
  const size_t rowbase = ((size_t)(b * LSEQ + c * CH) * NH + h) * 64;
  const size_t sbase = ((size_t)((b * NC + c) * NH + h)) * 4096;

  if (tid < 64) cs[tid] = A[(size_t)(b * LSEQ + c * CH + tid) * NH + h];

  for (int i = 0; i < 4; ++i) {
    int q = tid + i * 256;
    int r = q >> 4, f4 = (q & 15) * 4;
    *(v4f*)&Cm[r * LDSS + f4] = *(const v4f*)&Cp[rowbase + (size_t)r * rstride + f4];
    *(v4f*)&Sm[r * LDSS + f4] = *(const v4f*)&states[sbase + (size_t)r * 64 + f4];
  }
  __syncthreads();
  cumsum64(cs, tid);

  const int t0 = wv * 2;
  const int tr = (t0 >> 2) * 16;
  const int tc0 = (t0 & 3) * 16;
  const int tc1 = tc0 + 16;

  v8f acc0 = {}, acc1 = {};
  for (int kk = 0; kk < 64; kk += 4) {
    v2f a = ldA_rowmajor(Cm, tr, kk, half, l16);           // C[l, n]
    v2f b0 = ldB_transposed(Sm, tc0, kk, half, l16);       // S_prev^T[n, p]
    v2f b1 = ldB_transposed(Sm, tc1, kk, half, l16);
    acc0 = WMMA_F32(a, b0, acc0);
    acc1 = WMMA_F32(a, b1, acc1);
  }

  for (int j = 0; j < 8; ++j) {
    int row = tr + j + 8 * half;
    float sc = __expf(cs[row]);
    size_t o0 = rowbase + (size_t)row * rstride + tc0 + l16;
    size_t o1 = rowbase + (size_t)row * rstride + tc1 + l16;
    Y[o0] += sc * acc0[j];
    Y[o1] += sc * acc1[j];
  }
}

// =====================================================================
extern "C" void kernel_launch(void* const* d_in, const int* in_sizes, int n_in,
                              void* d_out, int out_size, void* d_ws, size_t ws_size,
                              hipStream_t stream) {
  const float* X = (const float*)d_in[0];
  const float* A = (const float*)d_in[1];
  const float* Bp = (const float*)d_in[2];
  const float* Cp = (const float*)d_in[3];
  float* Y = (float*)d_out;

  float* states = (float*)d_ws;                                  // B*NC*H*64*64 floats
  float* Ttot = states + (size_t)BSZ * NC * NH * 64 * 64;        // B*H*NC floats

  dim3 grid(BSZ * NC * NH);
  ssd_k1_intra<<<grid, 256, 0, stream>>>(X, A, Bp, Cp, Y, states, Ttot);
  ssd_k2_scan<<<dim3(BSZ * NH), 256, 0, stream>>>(states, Ttot);
  ssd_k3_off<<<grid, 256, 0, stream>>>(A, Cp, states, Y);
}